// Net_987842478551
// MI455X (gfx1250) — compile-verified
//
#include <hip/hip_runtime.h>

// GIN: 3x [scatter-add agg -> Linear -> BN -> ReLU -> Linear -> ReLU] -> pool -> MLP head
// Sizes: N=50000 nodes, E=800000 edges, feature dim H=128, G=512 graphs, OUT=64.
// Roofline: edge aggregation moves ~0.5 GB (gather + 102M f32 atomics) per pass,
// GEMMs total ~10 GFLOP -> memory/atomic bound. fp32 WMMA (v_wmma_f32_16x16x4_f32)
// keeps reference fp32 semantics at zero throughput cost versus bf16.

#define HDIM 128
#define OUTF 64
#define NGRAPH 512
#define NLAYER 3
#define BN_EPS 1e-5f

typedef __attribute__((ext_vector_type(2))) float v2f;
typedef __attribute__((ext_vector_type(8))) float v8f;

// ---------------- utility kernels ----------------

__global__ __launch_bounds__(256) void copy_f4(const float4* __restrict__ src,
                                               float4* __restrict__ dst, int n4) {
  int i = blockIdx.x * 256 + threadIdx.x;
  if (i < n4) dst[i] = src[i];
}

__global__ __launch_bounds__(256) void zero_f(float* __restrict__ p, int n) {
  int i = blockIdx.x * 256 + threadIdx.x;
  if (i < n) p[i] = 0.f;
}

// One wave32 per edge: 32 lanes x float4 = 128 floats per row.
// h[dst] += x[src]  (h pre-initialized to x, giving x + sum_{j->i} x_j)
__global__ __launch_bounds__(256) void edge_scatter(const float* __restrict__ x,
                                                    const int* __restrict__ src,
                                                    const int* __restrict__ dst,
                                                    float* __restrict__ h, int E) {
  int wid = (blockIdx.x * 256 + threadIdx.x) >> 5;
  int lane = threadIdx.x & 31;
  if (wid >= E) return;
  int s = src[wid];
  int d = dst[wid];
  float4 v = ((const float4*)(x + (size_t)s * HDIM))[lane];
  float* o = h + (size_t)d * HDIM + lane * 4;
  atomicAdd(o + 0, v.x);
  atomicAdd(o + 1, v.y);
  atomicAdd(o + 2, v.z);
  atomicAdd(o + 3, v.w);
}

// One wave32 per node: g[batch[n]] += x[n]
__global__ __launch_bounds__(256) void pool_scatter(const float* __restrict__ x,
                                                    const int* __restrict__ batch,
                                                    float* __restrict__ g, int N) {
  int wid = (blockIdx.x * 256 + threadIdx.x) >> 5;
  int lane = threadIdx.x & 31;
  if (wid >= N) return;
  int b = batch[wid];
  float4 v = ((const float4*)(x + (size_t)wid * HDIM))[lane];
  float* o = g + (size_t)b * HDIM + lane * 4;
  atomicAdd(o + 0, v.x);
  atomicAdd(o + 1, v.y);
  atomicAdd(o + 2, v.z);
  atomicAdd(o + 3, v.w);
}

// Column-wise sum and sum-of-squares of h1 [N x 128] -> stats[0:128]=sum, stats[128:256]=sumsq
__global__ __launch_bounds__(128) void col_stats(const float* __restrict__ h1,
                                                 float* __restrict__ stats, int N) {
  int c = threadIdx.x;
  float s = 0.f, s2 = 0.f;
  for (int r = blockIdx.x; r < N; r += gridDim.x) {
    float v = h1[(size_t)r * HDIM + c];
    s += v;
    s2 += v * v;
  }
  atomicAdd(&stats[c], s);
  atomicAdd(&stats[HDIM + c], s2);
}

// scale = gamma * rsqrt(var + eps); shift = beta - mu * scale
__global__ __launch_bounds__(128) void bn_finalize(const float* __restrict__ stats,
                                                   const float* __restrict__ gamma,
                                                   const float* __restrict__ beta,
                                                   float* __restrict__ scale,
                                                   float* __restrict__ shift, float invN) {
  int c = threadIdx.x;
  float mu = stats[c] * invN;
  float var = stats[HDIM + c] * invN - mu * mu;
  float sc = gamma[c] * rsqrtf(var + BN_EPS);
  scale[c] = sc;
  shift[c] = beta[c] - mu * sc;
}

// ---------------- WMMA GEMM ----------------
// Out[M x NCOLS] = op(A)[M x 128] @ W[128 x NCOLS] + bias, optional output ReLU.
// op(A) = relu(A*scale + shift) per-column (fused BN+ReLU) when BN == true.
//
// Compile-time NCOLS/BN/RELU -> fully unrolled, branch-free inner loops.
// One block = 256 threads = 8 waves; each wave owns one 16-row tile across all
// NCOLS columns (NCOLS/16 v8f accumulators). All 32 A fragments (64 floats/lane)
// are preloaded first (32 outstanding global_load_b64 -> one wait), then the main
// loop is a pure ds_load_b64 + v_wmma stream. W is staged in LDS transposed with
// a 4-per-column rotation so each B fragment is a conflict-free ds_load_b64.
//
// v_wmma_f32_16x16x4_f32 fragment layouts (wave32, ISA 7.12.2):
//   A 16x4: lane m (<16) holds A[m][k0],A[m][k0+1]; lane m+16 holds A[m][k0+2..3]
//   B 4x16: lane n (<16) holds B[k0][n],B[k0+1][n]; lane n+16 holds B[k0+2..3][n]
//   C/D: VGPR r, lanes 0-15 -> (M=r, N=lane); lanes 16-31 -> (M=r+8, N=lane-16)
template <int NCOLS, bool BN, bool RELU>
__global__ __launch_bounds__(256) void gemm128_wmma(
    const float* __restrict__ A, const float* __restrict__ W,
    const float* __restrict__ bias, float* __restrict__ Out,
    const float* __restrict__ in_scale, const float* __restrict__ in_shift,
    int M) {
  constexpr int NT = NCOLS / 16;          // column tiles per wave
  __shared__ float Wt[NCOLS * HDIM];      // <= 64 KB

  const int tid = threadIdx.x;
  // Stage W (row-major 128 x NCOLS) into LDS transposed + rotated (compile-time bounds).
#pragma unroll
  for (int it = 0; it < HDIM * NCOLS / 256; ++it) {
    int idx = it * 256 + tid;
    int k = idx / NCOLS;
    int n = idx % NCOLS;
    Wt[n * HDIM + ((k + 4 * (n & 15)) & (HDIM - 1))] = W[idx];
  }
  __syncthreads();

  const int lane = tid & 31;
  const int wave = tid >> 5;
  const int mBase = (blockIdx.x * 8 + wave) * 16;
  if (mBase < M) {  // wave-uniform: EXEC stays all-ones inside (WMMA requirement)
    const int half = lane >> 4;  // 0: K pair (k0,k0+1); 1: K pair (k0+2,k0+3)
    const int l15 = lane & 15;
    const float* __restrict__ Arow = A + (size_t)(mBase + l15) * HDIM;
    const int rot = 4 * l15;

    // Preload all 32 A fragments: 32 global_load_b64 in flight, then one wait.
    v2f af[32];
#pragma unroll
    for (int i = 0; i < 32; ++i) {
      const int kk = 4 * i + 2 * half;
      af[i].x = Arow[kk];
      af[i].y = Arow[kk + 1];
    }
    if constexpr (BN) {  // fused BatchNorm + ReLU on the GEMM input
#pragma unroll
      for (int i = 0; i < 32; ++i) {
        const int kk = 4 * i + 2 * half;
        af[i].x = fmaxf(af[i].x * in_scale[kk] + in_shift[kk], 0.f);
        af[i].y = fmaxf(af[i].y * in_scale[kk + 1] + in_shift[kk + 1], 0.f);
      }
    }

    v8f acc[NT] = {};
#pragma unroll
    for (int i = 0; i < 32; ++i) {
      const int kk = 4 * i + 2 * half;
#pragma unroll
      for (int j = 0; j < NT; ++j) {
        const int n = j * 16 + l15;
        // (kk + rot) is even, so the rotated pair never wraps -> contiguous b64
        const v2f b = *(const v2f*)&Wt[n * HDIM + ((kk + rot) & (HDIM - 1))];
        acc[j] = __builtin_amdgcn_wmma_f32_16x16x4_f32(
            false, af[i], false, b, (short)0, acc[j], false, false);
      }
    }

    // epilogue: bias (+ optional ReLU), direct stores
#pragma unroll
    for (int j = 0; j < NT; ++j) {
      const int n = j * 16 + l15;
      const float bv = bias[n];
#pragma unroll
      for (int r = 0; r < 8; ++r) {
        float v = acc[j][r] + bv;
        if constexpr (RELU) v = fmaxf(v, 0.f);
        Out[(size_t)(mBase + half * 8 + r) * NCOLS + n] = v;
      }
    }
  }
}

// ---------------- launch ----------------

extern "C" void kernel_launch(void* const* d_in, const int* in_sizes, int n_in,
                              void* d_out, int out_size, void* d_ws, size_t ws_size,
                              hipStream_t stream) {
  const float* x       = (const float*)d_in[0];   // [N,128]
  const float* conv_W1 = (const float*)d_in[1];   // [3,128,128]
  const float* conv_b1 = (const float*)d_in[2];   // [3,128]
  const float* conv_ga = (const float*)d_in[3];   // [3,128]
  const float* conv_be = (const float*)d_in[4];   // [3,128]
  const float* conv_W2 = (const float*)d_in[5];   // [3,128,128]
  const float* conv_b2 = (const float*)d_in[6];   // [3,128]
  const float* head_W1 = (const float*)d_in[7];   // [128,128]
  const float* head_b1 = (const float*)d_in[8];   // [128]
  const float* head_W2 = (const float*)d_in[9];   // [128,64]
  const float* head_b2 = (const float*)d_in[10];  // [64]
  const int*   edges   = (const int*)d_in[11];    // [2,E]
  const int*   batch   = (const int*)d_in[12];    // [N]
  float* out = (float*)d_out;                     // [512,64]

  const int N = in_sizes[0] / HDIM;
  const int E = in_sizes[11] / 2;
  const int* esrc = edges;
  const int* edst = edges + E;

  // workspace layout
  float* bufX  = (float*)d_ws;                   // N*128  (current x)
  float* bufH  = bufX  + (size_t)N * HDIM;       // N*128  (x + agg)
  float* bufH1 = bufH  + (size_t)N * HDIM;       // N*128  (post-Linear1)
  float* gbuf  = bufH1 + (size_t)N * HDIM;       // 512*128 (pooled)
  float* g1buf = gbuf  + (size_t)NGRAPH * HDIM;  // 512*128 (head hidden)
  float* stats = g1buf + (size_t)NGRAPH * HDIM;  // 256
  float* scale = stats + 2 * HDIM;               // 128
  float* shift = scale + HDIM;                   // 128

  const int n4      = N * HDIM / 4;
  const int cpBlks  = (n4 + 255) / 256;
  const int edBlks  = (E + 7) / 8;        // 8 waves (edges) per block
  const int plBlks  = (N + 7) / 8;        // 8 waves (nodes) per block
  const int gmBlksN = (N / 16 + 7) / 8;   // N % 16 == 0 (50000 = 3125*16)
  const int gmBlksG = (NGRAPH / 16 + 7) / 8;
  const float invN  = 1.0f / (float)N;

  const float* xcur = x;
  for (int l = 0; l < NLAYER; ++l) {
    // h = x + sum_{j->i} x_j
    copy_f4<<<cpBlks, 256, 0, stream>>>((const float4*)xcur, (float4*)bufH, n4);
    edge_scatter<<<edBlks, 256, 0, stream>>>(xcur, esrc, edst, bufH, E);
    // h1 = h @ W1 + b1
    gemm128_wmma<HDIM, false, false><<<gmBlksN, 256, 0, stream>>>(
        bufH, conv_W1 + (size_t)l * HDIM * HDIM, conv_b1 + l * HDIM, bufH1,
        nullptr, nullptr, N);
    // batch-norm stats over N rows
    zero_f<<<1, 256, 0, stream>>>(stats, 2 * HDIM);
    col_stats<<<256, 128, 0, stream>>>(bufH1, stats, N);
    bn_finalize<<<1, 128, 0, stream>>>(stats, conv_ga + l * HDIM, conv_be + l * HDIM,
                                       scale, shift, invN);
    // x_next = relu( relu(bn(h1)) @ W2 + b2 )   (BN+ReLU fused into A-load)
    gemm128_wmma<HDIM, true, true><<<gmBlksN, 256, 0, stream>>>(
        bufH1, conv_W2 + (size_t)l * HDIM * HDIM, conv_b2 + l * HDIM, bufX,
        scale, shift, N);
    xcur = bufX;
  }

  // global_add_pool -> g [512,128]
  zero_f<<<(NGRAPH * HDIM + 255) / 256, 256, 0, stream>>>(gbuf, NGRAPH * HDIM);
  pool_scatter<<<plBlks, 256, 0, stream>>>(bufX, batch, gbuf, N);

  // head MLP
  gemm128_wmma<HDIM, false, true><<<gmBlksG, 256, 0, stream>>>(
      gbuf, head_W1, head_b1, g1buf, nullptr, nullptr, NGRAPH);
  gemm128_wmma<OUTF, false, false><<<gmBlksG, 256, 0, stream>>>(
      g1buf, head_W2, head_b2, out, nullptr, nullptr, NGRAPH);
}